// R2N2_71021579206890
// MI455X (gfx1250) — compile-verified
//
#include <hip/hip_runtime.h>
#include <stdint.h>

// Problem constants (from reference): T=32 nodes, P=3 polarities, 17 relation mats.
#define T_NODES 32
#define P_DIM 3
#define DOC_F (T_NODES * P_DIM)     // 96 floats per document
#define DOCS_PER_BLOCK 128
#define NS_STRIDE 100               // padded LDS stride (floats); 16B-aligned, good bank spread
#define N_RELS_P1 17
#define CHUNKS_PER_THREAD 24        // (128 docs * 96 floats / 4) / 128 threads

// Builtin signature (from hipcc diagnostic): pointee is a 4-int vector.
typedef int v4i __attribute__((vector_size(16)));
typedef __attribute__((address_space(1))) v4i g1v4;   // global
typedef __attribute__((address_space(3))) v4i l3v4;   // LDS

// ---- CDNA5 async global->LDS copy (ASYNCcnt-tracked) --------------------
__device__ __forceinline__ void async_copy_b128(const void* gsrc, void* lds_dst) {
#if __has_builtin(__builtin_amdgcn_global_load_async_to_lds_b128)
    __builtin_amdgcn_global_load_async_to_lds_b128((g1v4*)gsrc, (l3v4*)lds_dst, 0, 0);
#else
    unsigned lds_off = (unsigned)(unsigned long long)(l3v4*)lds_dst;
    asm volatile("global_load_async_to_lds_b128 %0, %1, off"
                 :: "v"(lds_off), "v"((unsigned long long)gsrc)
                 : "memory");
#endif
}

__device__ __forceinline__ void wait_async0() {
#if __has_builtin(__builtin_amdgcn_s_wait_asynccnt)
    __builtin_amdgcn_s_wait_asynccnt(0);
#else
    asm volatile("s_wait_asynccnt 0" ::: "memory");
#endif
}

// tanh via one transcendental: tanh(x) = 1 - 2/(e^{2x}+1)
__device__ __forceinline__ float fast_tanh(float x) {
    float t = __builtin_amdgcn_exp2f(x * 2.8853900817779268f); // e^(2x)
    return 1.0f - 2.0f * __builtin_amdgcn_rcpf(t + 1.0f);
}

__global__ __launch_bounds__(DOCS_PER_BLOCK)
void r2n2_tree_kernel(const float* __restrict__ node_scores,
                      const int*   __restrict__ children,
                      const int*   __restrict__ rels,
                      const float* __restrict__ msg_scores,
                      const float* __restrict__ K,
                      const float* __restrict__ gamma_p,
                      float* __restrict__ out,
                      int B)
{
    __shared__ __align__(16) float  ns[DOCS_PER_BLOCK * NS_STRIDE];   // 51200 B
    __shared__ __align__(16) float4 krow[N_RELS_P1 * P_DIM];          // 816 B, padded K rows

    const int tid  = threadIdx.x;
    const int doc0 = blockIdx.x * DOCS_PER_BLOCK;

    // ---- Issue async staging of this block's node_scores FIRST (coalesced
    // b128; DMA runs while we stage K below). Chunk e4 = k*128 + tid maps to
    // local doc d = e4/24 and float4-within-doc j4 = e4%24; updated
    // incrementally (+5,+8 with wrap) to avoid per-chunk divisions.
    {
        int d  = tid / 24;
        int j4 = tid - d * 24;
        const float* gptr = node_scores + (size_t)doc0 * DOC_F + (size_t)tid * 4;
        #pragma unroll
        for (int k = 0; k < CHUNKS_PER_THREAD; ++k) {
            if (doc0 + d < B) {
                async_copy_b128(gptr, &ns[d * NS_STRIDE + j4 * 4]);
            }
            gptr += 128 * 4;          // next chunk for this thread (+512 floats)
            d += 5; j4 += 8;          // 128 chunks == 5 docs + 8 float4
            if (j4 >= 24) { j4 -= 24; d += 1; }
        }
    }

    // ---- Stage K into LDS as 16B rows while the async DMA is in flight ----
    for (int idx = tid; idx < N_RELS_P1 * P_DIM; idx += DOCS_PER_BLOCK) {
        int r = idx / P_DIM, p = idx % P_DIM;
        const float* kp = K + ((size_t)r * P_DIM + p) * P_DIM;
        krow[idx] = make_float4(kp[0], kp[1], kp[2], 0.0f);
    }

    wait_async0();
    __syncthreads();

    const int doc = doc0 + tid;
    if (doc >= B) return;

    float* myns = &ns[tid * NS_STRIDE];
    const int* cptr = children + (size_t)doc * DOC_F + P_DIM;  // start at node i=1
    const int* rptr = rels     + (size_t)doc * DOC_F + P_DIM;
    __builtin_prefetch(msg_scores + (size_t)doc * P_DIM, 0, 0);

#define ACCUM_CHILD(cidx, ridx)                                           \
    {                                                                     \
        int cb = (cidx) * 3;                                              \
        float s0 = myns[cb + 0], s1 = myns[cb + 1], s2 = myns[cb + 2];    \
        const float4 k0 = krow[(ridx) * 3 + 0];                           \
        const float4 k1 = krow[(ridx) * 3 + 1];                           \
        const float4 k2 = krow[(ridx) * 3 + 2];                           \
        a0 = fmaf(s0, k0.x, fmaf(s1, k1.x, fmaf(s2, k2.x, a0)));          \
        a1 = fmaf(s0, k0.y, fmaf(s1, k1.y, fmaf(s2, k2.y, a1)));          \
        a2 = fmaf(s0, k0.z, fmaf(s1, k1.z, fmaf(s2, k2.z, a2)));          \
    }

    // ---- Sequential tree-level sweep (thread-private: no barriers needed) ----
    for (int i = 1; i < T_NODES; ++i) {
        int c0 = cptr[0], c1 = cptr[1], c2 = cptr[2];
        int r0 = rptr[0], r1 = rptr[1], r2 = rptr[2];
        cptr += P_DIM; rptr += P_DIM;

        float a0 = 0.0f, a1 = 0.0f, a2 = 0.0f;
        ACCUM_CHILD(c0, r0)
        ACCUM_CHILD(c1, r1)
        ACCUM_CHILD(c2, r2)

        int ib = i * 3;
        myns[ib + 0] += fast_tanh(a0);
        myns[ib + 1] += fast_tanh(a1);
        myns[ib + 2] += fast_tanh(a2);
    }
#undef ACCUM_CHILD

    // ---- softmax(gamma * msg + ns[:, T-1]) ----
    const float g = *gamma_p;
    const float* mp = msg_scores + (size_t)doc * P_DIM;
    float x0 = fmaf(g, mp[0], myns[(T_NODES - 1) * 3 + 0]);
    float x1 = fmaf(g, mp[1], myns[(T_NODES - 1) * 3 + 1]);
    float x2 = fmaf(g, mp[2], myns[(T_NODES - 1) * 3 + 2]);
    float mx = fmaxf(x0, fmaxf(x1, x2));
    const float L2E = 1.4426950408889634f;
    float e0 = __builtin_amdgcn_exp2f((x0 - mx) * L2E);
    float e1 = __builtin_amdgcn_exp2f((x1 - mx) * L2E);
    float e2 = __builtin_amdgcn_exp2f((x2 - mx) * L2E);
    float inv = __builtin_amdgcn_rcpf(e0 + e1 + e2);
    float* op = out + (size_t)doc * P_DIM;
    op[0] = e0 * inv;
    op[1] = e1 * inv;
    op[2] = e2 * inv;
}

extern "C" void kernel_launch(void* const* d_in, const int* in_sizes, int n_in,
                              void* d_out, int out_size, void* d_ws, size_t ws_size,
                              hipStream_t stream) {
    const float* node_scores = (const float*)d_in[0];
    const int*   children    = (const int*)d_in[1];
    const int*   rels        = (const int*)d_in[2];
    const float* msg_scores  = (const float*)d_in[3];
    const float* K           = (const float*)d_in[4];
    const float* gamma_p     = (const float*)d_in[5];
    float* out = (float*)d_out;

    const int B = in_sizes[0] / DOC_F;                       // 65536
    const int grid = (B + DOCS_PER_BLOCK - 1) / DOCS_PER_BLOCK;

    r2n2_tree_kernel<<<dim3(grid), dim3(DOCS_PER_BLOCK), 0, stream>>>(
        node_scores, children, rels, msg_scores, K, gamma_p, out, B);
}